// VectorQuantizer_20409684591296
// MI455X (gfx1250) — compile-verified
//
#include <hip/hip_runtime.h>
#include <hip/hip_bf16.h>

typedef float v2f __attribute__((ext_vector_type(2)));
typedef float v8f __attribute__((ext_vector_type(8)));

#define R_TOT   65536        // N*H*W
#define D_DIM   256
#define K_CODE  1024
#define ROWS_PER_BLOCK 128   // 8 waves * 16 rows
#define NBLOCKS (R_TOT / ROWS_PER_BLOCK)   // 512

// ---------------------------------------------------------------------------
// Kernel 1: transpose codebook (D,K)->(K,D), compute ||e_k||^2, zero histogram
// ---------------------------------------------------------------------------
__global__ __launch_bounds__(256)
void vq_prep(const float* __restrict__ embed, float* __restrict__ embedT,
             float* __restrict__ enorm, int* __restrict__ counts)
{
    int g = blockIdx.x * 256 + threadIdx.x;          // 0 .. 65535
    #pragma unroll
    for (int s = 0; s < 4; ++s) {
        int idx = g + s * 65536;                     // 0 .. 262143
        int k = idx >> 8;
        int d = idx & 255;
        embedT[idx] = embed[(size_t)d * K_CODE + k]; // embedT[k][d]
    }
    if (g < K_CODE) {
        float sum = 0.f;
        for (int d = 0; d < D_DIM; ++d) {
            float e = embed[(size_t)d * K_CODE + g];
            sum += e * e;
        }
        enorm[g]  = sum;
        counts[g] = 0;
    }
}

// ---------------------------------------------------------------------------
// Kernel 2: fused distance-GEMM (f32 WMMA) + argmin + gather + diff + hist
// ---------------------------------------------------------------------------
__global__ __launch_bounds__(256)
void vq_main(const float* __restrict__ x, const float* __restrict__ embed,
             const float* __restrict__ embedT, const float* __restrict__ enorm,
             int* __restrict__ counts, float* __restrict__ partials,
             float* __restrict__ out_q, float* __restrict__ out_inds)
{
    // LDS: codebook tile as K-pairs -> one b64 per B fragment, conflict-free.
    __shared__ v2f   esp[2][D_DIM / 2][16];   // 32 KB
    __shared__ float en_lds[K_CODE];          // 4 KB
    __shared__ int   widx[8][16];
    __shared__ float wdiff[8];

    const int tid  = threadIdx.x;
    const int wave = tid >> 5;
    const int lane = tid & 31;
    const int m    = lane & 15;   // row within 16 (A) / col within 16 (B,C)
    const int kh   = lane >> 4;   // K-half selector per ISA f32 WMMA layout
    const int row0 = blockIdx.x * ROWS_PER_BLOCK + wave * 16;

    // enorm -> LDS
    for (int i = tid; i < K_CODE; i += 256) en_lds[i] = enorm[i];

    // Preload A: 16 rows x 256 (full K-depth) into 64 v2f fragments.
    // Lane (m,kh), fragment t holds x[row0+m][4t+2kh .. 4t+2kh+1]
    v2f a[64];
    const float* xrow = x + (size_t)(row0 + m) * D_DIM + 2 * kh;
    #pragma unroll
    for (int t = 0; t < 64; ++t)
        a[t] = *(const v2f*)(xrow + 4 * t);

    // Cooperative tile loaders (16 codebook columns x 256 depth, pair layout)
    v2f rbuf[8];
    auto load_tile_regs = [&](int ct) {
        #pragma unroll
        for (int s = 0; s < 8; ++s) {
            int idx = tid + s * 256;          // 0 .. 2047
            int dr  = idx >> 4;               // d-pair index 0..127
            int col = idx & 15;
            const float* p = embed + (size_t)(2 * dr) * K_CODE + ct * 16 + col;
            rbuf[s].x = p[0];
            rbuf[s].y = p[K_CODE];
        }
    };
    auto store_tile = [&](int buf) {
        #pragma unroll
        for (int s = 0; s < 8; ++s) {
            int idx = tid + s * 256;
            int dr  = idx >> 4;
            int col = idx & 15;
            esp[buf][dr][col] = rbuf[s];
        }
    };

    load_tile_regs(0);
    store_tile(0);
    __syncthreads();

    float minv[8];
    int   mini[8];
    #pragma unroll
    for (int j = 0; j < 8; ++j) { minv[j] = 3.4e38f; mini[j] = 0; }

    #pragma unroll 1
    for (int ct = 0; ct < 64; ++ct) {
        if (ct + 1 < 64) load_tile_regs(ct + 1);   // prefetch next tile
        const int cur = ct & 1;

        v8f c = {};
        #pragma unroll
        for (int t = 0; t < 64; ++t) {
            // B fragment: (K=4t+2kh, N=m) and (K=4t+2kh+1, N=m)
            v2f b = esp[cur][2 * t + kh][m];
            c = __builtin_amdgcn_wmma_f32_16x16x4_f32(
                    false, a[t], false, b, (short)0, c, false, false);
        }

        // distance (minus row-constant ||x||^2): enorm[k] - 2 * (x . e_k)
        const float en = en_lds[ct * 16 + m];
        #pragma unroll
        for (int j = 0; j < 8; ++j) {           // C VGPR j -> row j + 8*kh
            float s = en - 2.0f * c[j];
            if (s < minv[j]) { minv[j] = s; mini[j] = ct * 16 + m; }
        }

        __syncthreads();                         // done reading esp[cur] & old buf
        if (ct + 1 < 64) {
            store_tile((ct + 1) & 1);
            __syncthreads();
        }
    }

    // Cross-lane argmin within each 16-lane half (tie -> smaller index)
    #pragma unroll
    for (int j = 0; j < 8; ++j) {
        float v = minv[j];
        int   i = mini[j];
        #pragma unroll
        for (int off = 1; off < 16; off <<= 1) {
            float ov = __shfl_xor(v, off, 32);
            int   oi = __shfl_xor(i, off, 32);
            if (ov < v || (ov == v && oi < i)) { v = ov; i = oi; }
        }
        minv[j] = v; mini[j] = i;
    }

    // Lanes m==0 (lanes 0 and 16) own final results for rows j + 8*kh
    if (m == 0) {
        #pragma unroll
        for (int j = 0; j < 8; ++j) {
            int rloc = j + 8 * kh;
            int bi   = mini[j];
            widx[wave][rloc]      = bi;
            out_inds[row0 + rloc] = (float)bi;
            atomicAdd(&counts[bi], 1);
        }
    }
    __syncthreads();

    // Gather quantized row from L2-resident embedT, write output, diff accum.
    const int myidx = widx[wave][m];
    const float* qrow = embedT + (size_t)myidx * D_DIM + 2 * kh;
    float*       orow = out_q + (size_t)(row0 + m) * D_DIM + 2 * kh;
    float acc = 0.f;
    #pragma unroll
    for (int t = 0; t < 64; ++t) {
        v2f q = *(const v2f*)(qrow + 4 * t);
        *(v2f*)(orow + 4 * t) = q;
        float dx = q.x - a[t].x;
        float dy = q.y - a[t].y;
        acc += dx * dx + dy * dy;
    }
    #pragma unroll
    for (int off = 1; off < 32; off <<= 1)
        acc += __shfl_xor(acc, off, 32);
    if (lane == 0) wdiff[wave] = acc;
    __syncthreads();
    if (tid == 0) {
        float s = 0.f;
        #pragma unroll
        for (int w = 0; w < 8; ++w) s += wdiff[w];   // fixed order: deterministic
        partials[blockIdx.x] = s;
    }
}

// ---------------------------------------------------------------------------
// Kernel 3: perplexity from histogram + diff mean (fixed-order reductions)
// ---------------------------------------------------------------------------
__global__ __launch_bounds__(1024)
void vq_final(const int* __restrict__ counts, const float* __restrict__ partials,
              float* __restrict__ out_diff, float* __restrict__ out_perp)
{
    __shared__ float red[1024];
    const int t = threadIdx.x;

    float avg = (float)counts[t] * (1.0f / (float)R_TOT);
    red[t] = avg * logf(avg + 1e-10f);
    __syncthreads();
    for (int s = 512; s > 0; s >>= 1) {
        if (t < s) red[t] += red[t + s];
        __syncthreads();
    }
    const float H = -red[0];
    __syncthreads();

    red[t] = (t < NBLOCKS) ? partials[t] : 0.f;
    __syncthreads();
    for (int s = 512; s > 0; s >>= 1) {
        if (t < s) red[t] += red[t + s];
        __syncthreads();
    }
    if (t == 0) {
        *out_perp = expf(H);
        *out_diff = red[0] * (1.0f / ((float)R_TOT * (float)D_DIM));
    }
}

// ---------------------------------------------------------------------------
extern "C" void kernel_launch(void* const* d_in, const int* in_sizes, int n_in,
                              void* d_out, int out_size, void* d_ws, size_t ws_size,
                              hipStream_t stream)
{
    (void)in_sizes; (void)n_in; (void)out_size; (void)ws_size;

    const float* x     = (const float*)d_in[0];   // (64,32,32,256) f32
    const float* embed = (const float*)d_in[1];   // (256,1024) f32

    float* out      = (float*)d_out;
    float* out_q    = out;                         // 16777216 floats
    float* out_diff = out + 16777216;              // 1
    float* out_inds = out + 16777217;              // 65536
    float* out_perp = out + 16777217 + 65536;      // 1

    float* embedT   = (float*)d_ws;                // 262144 floats (1 MB)
    float* enorm    = embedT + 262144;             // 1024 floats
    int*   counts   = (int*)(enorm + 1024);        // 1024 ints
    float* partials = (float*)(counts + 1024);     // 512 floats

    vq_prep <<<256, 256, 0, stream>>>(embed, embedT, enorm, counts);
    vq_main <<<NBLOCKS, 256, 0, stream>>>(x, embed, embedT, enorm,
                                          counts, partials, out_q, out_inds);
    vq_final<<<1, 1024, 0, stream>>>(counts, partials, out_diff, out_perp);
}